// SPEEdges_47158740910668
// MI455X (gfx1250) — compile-verified
//
#include <hip/hip_runtime.h>
#include <hip/hip_bf16.h>

typedef __attribute__((ext_vector_type(2))) float v2f;
typedef __attribute__((ext_vector_type(8))) float v8f;

#define B_  8
#define N_  256
#define QD_ 64     // q*d = 4*16
#define K2_ 128    // 2*q*d
#define O_  64     // out_dim
#define C_  16
#define SROW 72    // padded LDS row stride (floats) to avoid bank conflicts

// ------------------------------------------------------------------
// Kernel 1: per-(b,q,d) MLP (1->32->32->16) folded with Wr -> WR/WI
//   WR[p*64+o] = sum_c w[p,c] * Wr[o, q*16+c]
//   WI[p*64+o] = sum_c w[p,c] * Wr[o, 64 + q*16+c]
// p = b*64 + q*16 + d  (matches eigenvals flat layout [b,q,d])
// ------------------------------------------------------------------
__global__ __launch_bounds__(64) void spe_prep_kernel(
    const float* __restrict__ eigenvals,
    const float* __restrict__ W1, const float* __restrict__ b1,
    const float* __restrict__ W2, const float* __restrict__ b2,
    const float* __restrict__ W3, const float* __restrict__ b3,
    const float* __restrict__ Wr,
    float* __restrict__ WRout, float* __restrict__ WIout)
{
    const int p = blockIdx.x;          // 0 .. B*QD-1
    const int o = threadIdx.x;         // 0 .. 63
    const int q = (p >> 4) & 3;

    const float x = eigenvals[p];

    float h1[32];
#pragma unroll
    for (int i = 0; i < 32; ++i) {
        float v = W1[i] * x + b1[i];
        h1[i] = v > 0.f ? v : 0.f;
    }
    float h2[32];
    for (int i = 0; i < 32; ++i) {
        float acc = b2[i];
#pragma unroll
        for (int j = 0; j < 32; ++j) acc += W2[i * 32 + j] * h1[j];
        h2[i] = acc > 0.f ? acc : 0.f;
    }

    float wr_acc = 0.f, wi_acc = 0.f;
    for (int c = 0; c < C_; ++c) {
        float wc = b3[c];
#pragma unroll
        for (int j = 0; j < 32; ++j) wc += W3[c * 32 + j] * h2[j];
        wr_acc += wc * Wr[o * K2_ + q * 16 + c];
        wi_acc += wc * Wr[o * K2_ + 64 + q * 16 + c];
    }
    WRout[p * O_ + o] = wr_acc;
    WIout[p * O_ + o] = wi_acc;
}

// ------------------------------------------------------------------
// Kernel 2: one workgroup (256 threads = 8 waves) per (b, m).
//   S[k][o] (k<64):  a_m[k]*WR - b_m[k]*WI       (pairs with X = re)
//   S[k][o] (k>=64): b_m[j]*WR + a_m[j]*WI, j=k-64 (pairs with X = im)
//   out[b, n, m, o] = sum_k X[b,n,k] * S[k][o] + br[o]
// GEMM [256 x 128] x [128 x 64] via V_WMMA_F32_16X16X4_F32.
// ------------------------------------------------------------------
__global__ __launch_bounds__(256) void spe_gram_kernel(
    const float* __restrict__ re, const float* __restrict__ im,
    const float* __restrict__ WR, const float* __restrict__ WI,
    const float* __restrict__ br,
    float* __restrict__ out)
{
    __shared__ float Ssh[K2_ * SROW];   // 128 * 72 * 4B = 36 KB

    const int bm = blockIdx.x;          // b*256 + m
    const int b  = bm >> 8;
    const int m  = bm & 255;
    const int t  = threadIdx.x;

    // ---- build S in LDS ----
    {
        const int o  = t & 63;
        const int kq = t >> 6;                       // 0..3
        const float* am  = re + (size_t)bm * QD_;
        const float* bmv = im + (size_t)bm * QD_;
        const float* wrB = WR + (size_t)b * QD_ * O_;
        const float* wiB = WI + (size_t)b * QD_ * O_;
#pragma unroll
        for (int i = 0; i < 32; ++i) {
            const int k = i * 4 + kq;
            const int j = k & 63;
            const float wr = wrB[j * O_ + o];
            const float wi = wiB[j * O_ + o];
            const float av = am[j];
            const float bv = bmv[j];
            Ssh[k * SROW + o] = (k < 64) ? (av * wr - bv * wi)
                                         : (bv * wr + av * wi);
        }
    }
    __syncthreads();

    // ---- WMMA GEMM ----
    const int wave  = t >> 5;
    const int lane  = t & 31;
    const int olane = lane & 15;
    const int hi    = lane >> 4;       // 0: lanes 0-15, 1: lanes 16-31
    const int row0  = wave * 32 + olane;   // n-tile pair nt = wave*2, wave*2+1
    const int row1  = row0 + 16;

    const v8f vzero = {0.f, 0.f, 0.f, 0.f, 0.f, 0.f, 0.f, 0.f};
    v8f acc[2][4];
#pragma unroll
    for (int i = 0; i < 2; ++i)
#pragma unroll
        for (int ot = 0; ot < 4; ++ot) acc[i][ot] = vzero;

    const float* xb_re = re + (size_t)b * N_ * QD_;
    const float* xb_im = im + (size_t)b * N_ * QD_;

#pragma unroll 4
    for (int ks = 0; ks < 32; ++ks) {
        const float* xp = (ks < 16) ? xb_re : xb_im;
        const int kof = ((ks * 4) & 63) + 2 * hi;        // A: K = kof, kof+1
        // A fragment (16x4 f32): VGPR0 = K even-pair base per half-wave
        v2f a0 = *(const v2f*)(xp + (size_t)row0 * QD_ + kof);
        v2f a1 = *(const v2f*)(xp + (size_t)row1 * QD_ + kof);

        const int krow = ks * 4 + 2 * hi;                // B: rows krow, krow+1
        const float* s0 = &Ssh[krow * SROW + olane];
        v2f bf[4];
#pragma unroll
        for (int ot = 0; ot < 4; ++ot) {
            bf[ot].x = s0[ot * 16];
            bf[ot].y = s0[ot * 16 + SROW];
        }
#pragma unroll
        for (int ot = 0; ot < 4; ++ot) {
            acc[0][ot] = __builtin_amdgcn_wmma_f32_16x16x4_f32(
                false, a0, false, bf[ot], (short)0, acc[0][ot], false, false);
            acc[1][ot] = __builtin_amdgcn_wmma_f32_16x16x4_f32(
                false, a1, false, bf[ot], (short)0, acc[1][ot], false, false);
        }
    }

    // ---- store D tiles: out[b][n][m][o], o contiguous (coalesced 64B) ----
#pragma unroll
    for (int ot = 0; ot < 4; ++ot) {
        const float brv = br[ot * 16 + olane];
#pragma unroll
        for (int i = 0; i < 2; ++i) {
            const int nbase = wave * 32 + i * 16 + 8 * hi;
#pragma unroll
            for (int r = 0; r < 8; ++r) {
                const int n = nbase + r;
                out[(((size_t)b * N_ + n) * N_ + m) * O_ + ot * 16 + olane] =
                    acc[i][ot][r] + brv;
            }
        }
    }
}

extern "C" void kernel_launch(void* const* d_in, const int* in_sizes, int n_in,
                              void* d_out, int out_size, void* d_ws, size_t ws_size,
                              hipStream_t stream) {
    const float* eigenvals = (const float*)d_in[0];
    const float* re = (const float*)d_in[1];
    const float* im = (const float*)d_in[2];
    // d_in[3] = mask : unused by the reference computation
    const float* W1 = (const float*)d_in[4];
    const float* b1 = (const float*)d_in[5];
    const float* W2 = (const float*)d_in[6];
    const float* b2 = (const float*)d_in[7];
    const float* W3 = (const float*)d_in[8];
    const float* b3 = (const float*)d_in[9];
    const float* Wr = (const float*)d_in[10];
    const float* br = (const float*)d_in[11];
    float* out = (float*)d_out;

    float* WRw = (float*)d_ws;               // B*QD*O floats (128 KB)
    float* WIw = WRw + B_ * QD_ * O_;        // another 128 KB

    spe_prep_kernel<<<B_ * QD_, 64, 0, stream>>>(
        eigenvals, W1, b1, W2, b2, W3, b3, Wr, WRw, WIw);
    spe_gram_kernel<<<B_ * N_, 256, 0, stream>>>(
        re, im, WRw, WIw, br, out);
}